// GCN_4_3_relu_8332236554731
// MI455X (gfx1250) — compile-verified
//
#include <hip/hip_runtime.h>
#include <math.h>

#define F 300
#define FC4 (F / 4)          // 75 float4 chunks per row
#define TILES_N 19           // ceil(300/16)

typedef __attribute__((ext_vector_type(2))) float v2f;
typedef __attribute__((ext_vector_type(8))) float v8f;

// ---------------------------------------------------------------- zero (float4)
__global__ void zero_f4(float4* __restrict__ p, long long n4) {
    long long i = (long long)blockIdx.x * blockDim.x + threadIdx.x;
    if (i < n4) {
        float4 z; z.x = 0.f; z.y = 0.f; z.z = 0.f; z.w = 0.f;
        p[i] = z;
    }
}

// ------------------------------------------------- edge scatter-add aggregation
// thread = (edge, 4-float chunk). Consecutive threads cover consecutive chunks
// of the same edge -> coalesced float4 gather from src row, coalesced
// global_atomic_add_f32 into dst row.
__global__ void scatter_add(const float* __restrict__ x, const int* __restrict__ src,
                            const int* __restrict__ dst, float* __restrict__ agg, int E) {
    long long idx = (long long)blockIdx.x * blockDim.x + threadIdx.x;
    long long total = (long long)E * FC4;
    if (idx >= total) return;
    int e = (int)(idx / FC4);
    int c = (int)(idx % FC4);
    int s = src[e];
    int d = dst[e];
    const float4 v = *(const float4*)(x + (long long)s * F + c * 4);
    float* dp = agg + (long long)d * F + c * 4;
    atomicAdd(dp + 0, v.x);
    atomicAdd(dp + 1, v.y);
    atomicAdd(dp + 2, v.z);
    atomicAdd(dp + 3, v.w);
}

// ------------------------------------------------------------- WMMA fp32 GEMM
// out[m, n] = sum_k A[m,k] * W[n,k] + bias[n]   (A: nrows x 300, W: 300 x 300)
// One wave32 computes one 16x16 output tile with V_WMMA_F32_16X16X4_F32,
// stepping K by 4 (75 WMMAs). EXEC stays all-1s for every WMMA.
__global__ void gemm_wmma(const float* __restrict__ A, const float* __restrict__ Wt,
                          const float* __restrict__ bias, float* __restrict__ out,
                          int mtiles, int nrows) {
    int wave = blockIdx.x * (blockDim.x >> 5) + (threadIdx.x >> 5);
    if (wave >= mtiles * TILES_N) return;          // whole-wave exit only
    int lane = threadIdx.x & 31;
    int l16  = lane & 15;
    int half = lane >> 4;                          // 0: K=0,1   1: K=2,3

    int tm = wave / TILES_N;
    int tn = wave % TILES_N;

    int arow = tm * 16 + l16;
    if (arow > nrows - 1) arow = nrows - 1;        // clamp (N=100000 is exact anyway)
    int bcol = tn * 16 + l16;
    int bcolc = (bcol < F) ? bcol : (F - 1);       // ragged N tile: clamp column;
                                                   // garbage cols are never stored
    const float* ap = A  + (long long)arow  * F + 2 * half;
    const float* bp = Wt + (long long)bcolc * F + 2 * half;

    v8f c = {};
    #pragma unroll 5
    for (int k = 0; k < F; k += 4) {
        v2f a = *(const v2f*)(ap + k);             // A: lane row, K = k+2*half .. +1
        v2f b = *(const v2f*)(bp + k);             // B: lane col, K = k+2*half .. +1
        c = __builtin_amdgcn_wmma_f32_16x16x4_f32(
                false, a, false, b, (short)0, c, false, false);
    }

    if (bcol < F) {
        float bv = bias[bcol];
        #pragma unroll
        for (int r = 0; r < 8; ++r) {              // D: VGPR r -> row r+8*half, col l16
            int orow = tm * 16 + half * 8 + r;
            if (orow < nrows)
                out[(long long)orow * F + bcol] = c[r] + bv;
        }
    }
}

// ----------------------------------------------------- wave32-per-row softmax
__global__ void softmax_rows(float* __restrict__ x, int nrows) {
    int wave = blockIdx.x * (blockDim.x >> 5) + (threadIdx.x >> 5);
    int lane = threadIdx.x & 31;
    if (wave >= nrows) return;
    float* row = x + (long long)wave * F;

    float v[10];
    float m = -INFINITY;
    #pragma unroll
    for (int i = 0; i < 10; ++i) {
        int j = lane + 32 * i;
        v[i] = (j < F) ? row[j] : -INFINITY;
        m = fmaxf(m, v[i]);
    }
    #pragma unroll
    for (int off = 16; off > 0; off >>= 1) m = fmaxf(m, __shfl_xor(m, off, 32));

    float s = 0.f;
    #pragma unroll
    for (int i = 0; i < 10; ++i) { v[i] = expf(v[i] - m); s += v[i]; } // exp(-inf)=0
    #pragma unroll
    for (int off = 16; off > 0; off >>= 1) s += __shfl_xor(s, off, 32);

    float inv = 1.f / s;
    #pragma unroll
    for (int i = 0; i < 10; ++i) {
        int j = lane + 32 * i;
        if (j < F) row[j] = v[i] * inv;
    }
}

// ------------------------------------------------------------------ relu
__global__ void relu_k(float* __restrict__ x, long long n) {
    long long i = (long long)blockIdx.x * blockDim.x + threadIdx.x;
    if (i < n) x[i] = fmaxf(x[i], 0.f);
}

// -------------------------------- final 300->3 layer + log_softmax, thread/node
__global__ void final_layer(const float* __restrict__ agg, const float* __restrict__ W4,
                            const float* __restrict__ b4, float* __restrict__ out,
                            int nrows) {
    int n = blockIdx.x * blockDim.x + threadIdx.x;
    if (n >= nrows) return;
    const float* a = agg + (long long)n * F;
    float a0 = b4[0], a1 = b4[1], a2 = b4[2];
    for (int k = 0; k < F; ++k) {
        float av = a[k];                      // sequential row walk, line-reuse in L0
        a0 = fmaf(av, W4[k],         a0);     // W4 rows are wave-uniform (cached)
        a1 = fmaf(av, W4[F + k],     a1);
        a2 = fmaf(av, W4[2 * F + k], a2);
    }
    float m  = fmaxf(a0, fmaxf(a1, a2));
    float ls = logf(expf(a0 - m) + expf(a1 - m) + expf(a2 - m));
    out[(long long)n * 3 + 0] = a0 - m - ls;
    out[(long long)n * 3 + 1] = a1 - m - ls;
    out[(long long)n * 3 + 2] = a2 - m - ls;
}

extern "C" void kernel_launch(void* const* d_in, const int* in_sizes, int n_in,
                              void* d_out, int out_size, void* d_ws, size_t ws_size,
                              hipStream_t stream) {
    const float* feat = (const float*)d_in[0];
    const int*   src  = (const int*)d_in[1];
    const int*   dst  = (const int*)d_in[2];
    const float* W1 = (const float*)d_in[3]; const float* b1 = (const float*)d_in[4];
    const float* W2 = (const float*)d_in[5]; const float* b2 = (const float*)d_in[6];
    const float* W3 = (const float*)d_in[7]; const float* b3 = (const float*)d_in[8];
    const float* W4 = (const float*)d_in[9]; const float* b4 = (const float*)d_in[10];

    const int nnodes = in_sizes[0] / F;     // 100000
    const int E      = in_sizes[1];         // 800000
    const long long nelem = (long long)nnodes * F;

    float* agg = (float*)d_ws;              // [nnodes, 300]
    float* xb  = agg + nelem;               // [nnodes, 300]
    float* outp = (float*)d_out;            // [nnodes, 3]

    const int mtiles = (nnodes + 15) / 16;
    const int tiles  = mtiles * TILES_N;

    const int zb  = (int)((nelem / 4 + 255) / 256);
    const int scb = (int)(((long long)E * FC4 + 255) / 256);
    const int gb  = (tiles + 7) / 8;        // 8 waves / 256-thread block
    const int smb = (nnodes + 7) / 8;
    const int rb  = (int)((nelem + 255) / 256);
    const int fb  = (nnodes + 255) / 256;

    // layer 1: softmax(agg(feat) @ W1.T + b1)
    zero_f4   <<<zb,  256, 0, stream>>>((float4*)agg, nelem / 4);
    scatter_add<<<scb, 256, 0, stream>>>(feat, src, dst, agg, E);
    gemm_wmma <<<gb,  256, 0, stream>>>(agg, W1, b1, xb, mtiles, nnodes);
    softmax_rows<<<smb, 256, 0, stream>>>(xb, nnodes);

    // layer 2
    zero_f4   <<<zb,  256, 0, stream>>>((float4*)agg, nelem / 4);
    scatter_add<<<scb, 256, 0, stream>>>(xb, src, dst, agg, E);
    gemm_wmma <<<gb,  256, 0, stream>>>(agg, W2, b2, xb, mtiles, nnodes);
    softmax_rows<<<smb, 256, 0, stream>>>(xb, nnodes);

    // layer 3: relu
    zero_f4   <<<zb,  256, 0, stream>>>((float4*)agg, nelem / 4);
    scatter_add<<<scb, 256, 0, stream>>>(xb, src, dst, agg, E);
    gemm_wmma <<<gb,  256, 0, stream>>>(agg, W3, b3, xb, mtiles, nnodes);
    relu_k    <<<rb,  256, 0, stream>>>(xb, nelem);

    // layer 4: log_softmax(agg @ W4.T + b4)
    zero_f4   <<<zb,  256, 0, stream>>>((float4*)agg, nelem / 4);
    scatter_add<<<scb, 256, 0, stream>>>(xb, src, dst, agg, E);
    final_layer<<<fb, 256, 0, stream>>>(agg, W4, b4, outp, nnodes);
}